// LogicConv2d_65644280152692
// MI455X (gfx1250) — compile-verified
//
#include <hip/hip_runtime.h>

typedef __attribute__((ext_vector_type(2))) float v2f;
typedef __attribute__((ext_vector_type(2), aligned(4))) float v2fu;  // 4B-aligned b64 loads
typedef __attribute__((ext_vector_type(4))) float v4f;
typedef __attribute__((ext_vector_type(8))) float v8f;

#define NUNITS 81
// float offsets into d_ws (all regions 16B-aligned)
#define WA_OFF    0      // 81*9   softmax(sel_a)
#define WB_OFF    736    // 81*9   softmax(sel_b)
#define WMIX_OFF  1472   // 81*4   per-unit mix coefficients {c0,cA,cB,cAB}
#define BFRAG_OFF 2768   // 9*2*3*32*2 pre-swizzled WMMA B fragments
#define NFRAG     (9 * 2 * 3 * 32)   // v2f entries: ((p*2+t)*3+kk)*32+lane

// Hardware-K -> logical patch element (permuted so A loads are contiguous
// b64 pairs per lane half; -1 = padding slot, B row forced to zero).
__device__ __constant__ int c_hw2i[12] = {0, 1, 3, 4, 6, 7, 2, -1, 5, -1, -1, 8};

// ---------------------------------------------------------------------------
// Prep (one block, runs once):
//  * softmax(sel_a), softmax(sel_b) into WA/WB
//  * softmax(op) folded into 4 affine mix coefficients per unit:
//      r = c0 + cA*a + cB*b + cAB*a*b   (exact expansion of the 14 terms)
//  * B fragments pre-swizzled into per-lane WMMA layout
//    (B 4x16 f32: vgpr0 = {K=4kk | K=4kk+2}, vgpr1 = {K=4kk+1 | K=4kk+3},
//     N = lane%16; zeros baked in for col>=9 and padding K slots)
// ---------------------------------------------------------------------------
__global__ void logic_prep_kernel(const float* __restrict__ sa,
                                  const float* __restrict__ sb,
                                  const float* __restrict__ op,
                                  float* __restrict__ ws) {
  const int tid = threadIdx.x;
  if (tid < NUNITS) {
    {
      const float* src = sa + tid * 9;
      float m = src[0];
#pragma unroll
      for (int i = 1; i < 9; ++i) m = fmaxf(m, src[i]);
      float e[9], s = 0.f;
#pragma unroll
      for (int i = 0; i < 9; ++i) { e[i] = expf(src[i] - m); s += e[i]; }
      const float inv = 1.0f / s;
#pragma unroll
      for (int i = 0; i < 9; ++i) ws[WA_OFF + tid * 9 + i] = e[i] * inv;
    }
    {
      const float* src = sb + tid * 9;
      float m = src[0];
#pragma unroll
      for (int i = 1; i < 9; ++i) m = fmaxf(m, src[i]);
      float e[9], s = 0.f;
#pragma unroll
      for (int i = 0; i < 9; ++i) { e[i] = expf(src[i] - m); s += e[i]; }
      const float inv = 1.0f / s;
#pragma unroll
      for (int i = 0; i < 9; ++i) ws[WB_OFF + tid * 9 + i] = e[i] * inv;
    }
    {
      const float* src = op + tid * 16;
      float m = src[0];
#pragma unroll
      for (int i = 1; i < 16; ++i) m = fmaxf(m, src[i]);
      float w[16], s = 0.f;
#pragma unroll
      for (int i = 0; i < 16; ++i) { w[i] = expf(src[i] - m); s += w[i]; }
      const float inv = 1.0f / s;
#pragma unroll
      for (int i = 0; i < 16; ++i) w[i] *= inv;
      const float c0  = w[8] + w[9] + w[10] + w[11] + w[12] + w[13] + w[14] + w[15];
      const float cA  = w[2] + w[3] + w[6] + w[7] - w[8] - w[9] - w[12] - w[13];
      const float cB  = w[4] + w[5] + w[6] + w[7] - w[8] - w[9] - w[10] - w[11];
      const float cAB = w[1] - w[2] - w[4] - 2.0f * w[6] - w[7] + w[8] +
                        2.0f * w[9] + w[11] + w[13] - w[14];
      ws[WMIX_OFF + tid * 4 + 0] = c0;
      ws[WMIX_OFF + tid * 4 + 1] = cA;
      ws[WMIX_OFF + tid * 4 + 2] = cB;
      ws[WMIX_OFF + tid * 4 + 3] = cAB;
    }
  }
  __syncthreads();
  for (int idx = tid; idx < NFRAG; idx += blockDim.x) {
    const int lane  = idx & 31;
    const int rest  = idx >> 5;       // ((p*2+t)*3+kk)
    const int kk    = rest % 3;
    const int pt    = rest / 3;       // p*2+t
    const int t     = pt & 1;
    const int p     = pt >> 1;
    const int col   = lane & 15;
    const int khalf = (lane >> 4) << 1;
    const float* tbl = ws + (t ? WB_OFF : WA_OFF);
    const int k0 = 4 * kk + khalf;
    const int i0 = c_hw2i[k0];
    const int i1 = c_hw2i[k0 + 1];
    float v0 = 0.f, v1 = 0.f;
    if (col < 9) {
      if (i0 >= 0) v0 = tbl[(9 * p + col) * 9 + i0];
      if (i1 >= 0) v1 = tbl[(9 * p + col) * 9 + i1];
    }
    ws[BFRAG_OFF + idx * 2 + 0] = v0;
    ws[BFRAG_OFF + idx * 2 + 1] = v1;
  }
}

// ---------------------------------------------------------------------------
// Main: one wave = 16 batch rows. Per patch p:
//   C_a = A(16x12) * Ba(12x16),  C_b = A(16x12) * Bb(12x16)
// via 6x V_WMMA_F32_16X16X4_F32.
// A loads: three b64 per lane at permuted offsets; "extra" elements read in
// each pair are in-bounds and hit zero B rows, so they contribute nothing.
//   lanes 0-15  (K = 4kk,4kk+1): offsets {0,1} {18,19} {11,12*}
//   lanes 16-31 (K = 4kk+2,+3) : offsets {9,10} {2,3*}  {19*,20}   (* = x0)
// Epilogue: r = c0 + cA*a + cB*b + cAB*a*b  (1 mul + 3 FMA per output).
// All loads/stores use per-lane base pointers + compile-time immediate
// offsets, so the unrolled loop carries no address arithmetic.
// ---------------------------------------------------------------------------
__global__ void __launch_bounds__(128, 1)
logic_wmma_kernel(const float* __restrict__ x,
                  const float* __restrict__ ws,
                  float* __restrict__ out,
                  int nBatch) {
  const int lane   = threadIdx.x & 31;
  const int waveId = threadIdx.x >> 5;
  const long base  = ((long)blockIdx.x * 4 + waveId) * 16;
  if (base + 16 > nBatch) return;          // wave-uniform: EXEC stays all-1s

  const int col  = lane & 15;              // N column / A row within tile
  const bool hiM = (lane >= 16);           // C layout: rows 8..15

  // Per-lane base pointers (computed once; loop uses immediate offsets only)
  const float* xrow = x + (base + col) * 81;
  const float* xr0  = xrow + (hiM ? 9 : 0);
  const float* xr1  = xrow + (hiM ? 2 : 18);
  const float* xr2  = xrow + (hiM ? 19 : 11);
  const v2f*   bp   = (const v2f*)(ws + BFRAG_OFF) + lane;
  const float* wc   = ws + WMIX_OFF + col * 4;
  float*       outp = out + (base + (hiM ? 8 : 0)) * 81 + col;
  const bool   act  = (col < 9);

#pragma unroll
  for (int p = 0; p < 9; ++p) {
    const int pbase = (p / 3) * 27 + (p % 3) * 3;  // patch origin in 9x9 img

    const v2f a0 = *(const v2fu*)(xr0 + pbase);
    const v2f a1 = *(const v2fu*)(xr1 + pbase);
    const v2f a2 = *(const v2fu*)(xr2 + pbase);

    const v2f ba0 = bp[(p * 6 + 0) * 32];
    const v2f ba1 = bp[(p * 6 + 1) * 32];
    const v2f ba2 = bp[(p * 6 + 2) * 32];
    const v2f bb0 = bp[(p * 6 + 3) * 32];
    const v2f bb1 = bp[(p * 6 + 4) * 32];
    const v2f bb2 = bp[(p * 6 + 5) * 32];

    v8f cAcc = {};
    v8f cBcc = {};
    cAcc = __builtin_amdgcn_wmma_f32_16x16x4_f32(false, a0, false, ba0, (short)0, cAcc, false, false);
    cAcc = __builtin_amdgcn_wmma_f32_16x16x4_f32(false, a1, false, ba1, (short)0, cAcc, false, false);
    cAcc = __builtin_amdgcn_wmma_f32_16x16x4_f32(false, a2, false, ba2, (short)0, cAcc, false, false);
    cBcc = __builtin_amdgcn_wmma_f32_16x16x4_f32(false, a0, false, bb0, (short)0, cBcc, false, false);
    cBcc = __builtin_amdgcn_wmma_f32_16x16x4_f32(false, a1, false, bb1, (short)0, cBcc, false, false);
    cBcc = __builtin_amdgcn_wmma_f32_16x16x4_f32(false, a2, false, bb2, (short)0, cBcc, false, false);

    // Epilogue: lane owns unit u = 9p+col (col<9), batch rows m(+8 if hiM)
    if (act) {
      const v4f c = *(const v4f*)(wc + 36 * p);  // {c0,cA,cB,cAB}, 16B aligned
#pragma unroll
      for (int m = 0; m < 8; ++m) {
        const float av = cAcc[m];
        const float bv = cBcc[m];
        const float r  = fmaf(av * bv, c.w, fmaf(bv, c.z, fmaf(av, c.y, c.x)));
        __builtin_nontemporal_store(r, outp + 9 * p + 81 * m);  // const offsets
      }
    }
  }
}

// ---------------------------------------------------------------------------
extern "C" void kernel_launch(void* const* d_in, const int* in_sizes, int n_in,
                              void* d_out, int out_size, void* d_ws, size_t ws_size,
                              hipStream_t stream) {
  const float* x  = (const float*)d_in[0];
  const float* sa = (const float*)d_in[1];
  const float* sb = (const float*)d_in[2];
  const float* op = (const float*)d_in[3];
  float* out = (float*)d_out;
  float* ws  = (float*)d_ws;

  const int nBatch = in_sizes[0] / 81;     // (B,1,9,9) flat

  logic_prep_kernel<<<1, 128, 0, stream>>>(sa, sb, op, ws);

  const int rowsPerBlock = 64;             // 4 waves * 16 batch rows
  const int nBlocks = (nBatch + rowsPerBlock - 1) / rowsPerBlock;
  logic_wmma_kernel<<<nBlocks, 128, 0, stream>>>(x, ws, out, nBatch);
}